// GNN_18090402251169
// MI455X (gfx1250) — compile-verified
//
#include <hip/hip_runtime.h>

#define NN 50000
#define NE 800000
#define DIN 96
#define HID 128
#define NG 256

typedef __attribute__((ext_vector_type(2))) float v2f;
typedef __attribute__((ext_vector_type(8))) float v8f;

// Device-scope fp32 atomic add (L2 atomic units); avoids CAS-loop lowering.
__device__ __forceinline__ void gatom_add_f32(float* p, float v) {
  asm volatile("global_atomic_add_f32 %0, %1, off scope:SCOPE_DEV"
               :: "v"(p), "v"(v) : "memory");
}

// agg[dst] += x[src] * w  for each edge, F floats per row, float4 chunks.
template <int F>
__global__ void scatter_edges(const float* __restrict__ h,
                              const int* __restrict__ ei,
                              const float* __restrict__ w,
                              float* __restrict__ agg) {
  constexpr int C = F / 4;
  int tid = blockIdx.x * blockDim.x + threadIdx.x;
  if (tid >= NE * C) return;
  int e = tid / C;
  int c = tid % C;
  int s = ei[e];
  int d = ei[NE + e];
  float we = w[e];
  const float4 xv = *(const float4*)(h + (long long)s * F + c * 4);
  float* dp = agg + (long long)d * F + c * 4;
  gatom_add_f32(dp + 0, xv.x * we);
  gatom_add_f32(dp + 1, xv.y * we);
  gatom_add_f32(dp + 2, xv.z * we);
  gatom_add_f32(dp + 3, xv.w * we);
}

// out = [relu](Aagg @ Wrel^T + bias + Aroot @ Wroot^T)
// One wave computes one 16x16 tile via V_WMMA_F32_16X16X4_F32.
// If POOL: instead of dense store, atomically add into pool[batch[row]][col].
template <int K, bool RELU, bool POOL>
__global__ void graphconv_gemm(const float* __restrict__ Aagg,
                               const float* __restrict__ Aroot,
                               const float* __restrict__ Wrel,   // [HID,K] row-major
                               const float* __restrict__ Wroot,  // [HID,K] row-major
                               const float* __restrict__ bias,   // [HID]
                               const int* __restrict__ batch,    // [NN] (POOL only)
                               float* __restrict__ out) {        // [NN,HID] or [NG,HID]
  int wave = blockIdx.x * (blockDim.x >> 5) + (threadIdx.x >> 5);
  constexpr int NT = HID / 16;  // 8 column tiles
  int mt = wave / NT, nt = wave % NT;
  int lane = threadIdx.x & 31;
  int half = lane >> 4;  // 0: lanes 0-15, 1: lanes 16-31
  int r = lane & 15;
  int m0 = mt * 16, n0 = nt * 16;

  // ISA 7.12.2: A 16x4 f32 -> lane(r) VGPR0/1 = A[r][2*half + {0,1}]
  //             B 4x16 f32 -> lane(r) VGPR0/1 = B[2*half + {0,1}][r] = W[n0+r][k..]
  const float* arow = Aagg + (long long)(m0 + r) * K;
  const float* rrow = Aroot + (long long)(m0 + r) * K;
  const float* wrel = Wrel + (long long)(n0 + r) * K;
  const float* wroo = Wroot + (long long)(n0 + r) * K;

  v8f c = {};
#pragma unroll 4
  for (int k = 0; k < K; k += 4) {
    int ko = k + 2 * half;
    v2f a, b;
    a.x = arow[ko]; a.y = arow[ko + 1];
    b.x = wrel[ko]; b.y = wrel[ko + 1];
    c = __builtin_amdgcn_wmma_f32_16x16x4_f32(false, a, false, b, (short)0, c,
                                              false, false);
  }
#pragma unroll 4
  for (int k = 0; k < K; k += 4) {
    int ko = k + 2 * half;
    v2f a, b;
    a.x = rrow[ko]; a.y = rrow[ko + 1];
    b.x = wroo[ko]; b.y = wroo[ko + 1];
    c = __builtin_amdgcn_wmma_f32_16x16x4_f32(false, a, false, b, (short)0, c,
                                              false, false);
  }

  float bv = bias[n0 + r];
#pragma unroll
  for (int j = 0; j < 8; ++j) {
    int row = m0 + j + 8 * half;  // C/D: VGPR j -> M = j (lanes 0-15) / j+8
    float v = c[j] + bv;
    if (RELU) v = fmaxf(v, 0.0f);
    if (POOL) {
      int g = batch[row];
      gatom_add_f32(out + (long long)g * HID + n0 + r, v);
    } else {
      out[(long long)row * HID + n0 + r] = v;
    }
  }
}

__global__ void count_nodes(const int* __restrict__ batch, int* __restrict__ cnt) {
  int i = blockIdx.x * blockDim.x + threadIdx.x;
  if (i < NN) atomicAdd(&cnt[batch[i]], 1);
}

// out[g] = relu( dot(pool[g]/max(cnt,1), Wlin) + blin ), one wave per graph.
__global__ void pool_linear(const float* __restrict__ pool,
                            const int* __restrict__ cnt,
                            const float* __restrict__ Wlin,
                            const float* __restrict__ blin,
                            float* __restrict__ out) {
  int g = blockIdx.x * (blockDim.x >> 5) + (threadIdx.x >> 5);
  int lane = threadIdx.x & 31;
  if (g >= NG) return;
  float s = 0.0f;
#pragma unroll
  for (int f = lane; f < HID; f += 32) s += pool[g * HID + f] * Wlin[f];
#pragma unroll
  for (int off = 16; off; off >>= 1) s += __shfl_xor(s, off, 32);
  if (lane == 0) {
    float c = fmaxf((float)cnt[g], 1.0f);
    out[g] = fmaxf(s / c + blin[0], 0.0f);
  }
}

extern "C" void kernel_launch(void* const* d_in, const int* in_sizes, int n_in,
                              void* d_out, int out_size, void* d_ws, size_t ws_size,
                              hipStream_t stream) {
  const float* x     = (const float*)d_in[0];
  const int*   ei    = (const int*)d_in[1];
  const int*   batch = (const int*)d_in[2];
  const float* ea    = (const float*)d_in[3];
  const float* Wrel1 = (const float*)d_in[4];
  const float* brel1 = (const float*)d_in[5];
  const float* Wroot1= (const float*)d_in[6];
  const float* Wrel3 = (const float*)d_in[7];
  const float* brel3 = (const float*)d_in[8];
  const float* Wroot3= (const float*)d_in[9];
  const float* Wlin  = (const float*)d_in[10];
  const float* blin  = (const float*)d_in[11];
  float* out = (float*)d_out;

  char* ws = (char*)d_ws;
  size_t off = 0;
  auto carve = [&](size_t bytes) -> void* {
    void* p = ws + off;
    off += (bytes + 255) & ~(size_t)255;
    return p;
  };
  float* agg1 = (float*)carve(sizeof(float) * (size_t)NN * DIN);
  float* h1   = (float*)carve(sizeof(float) * (size_t)NN * HID);
  float* agg2 = (float*)carve(sizeof(float) * (size_t)NN * HID);
  float* pool = (float*)carve(sizeof(float) * (size_t)NG * HID);
  int*   cnt  = (int*)carve(sizeof(int) * (size_t)NG);

  hipMemsetAsync(agg1, 0, sizeof(float) * (size_t)NN * DIN, stream);
  hipMemsetAsync(agg2, 0, sizeof(float) * (size_t)NN * HID, stream);
  hipMemsetAsync(pool, 0, sizeof(float) * (size_t)NG * HID, stream);
  hipMemsetAsync(cnt, 0, sizeof(int) * (size_t)NG, stream);

  // conv1 scatter: NE * 24 chunk-threads
  {
    int total = NE * (DIN / 4);
    scatter_edges<DIN><<<(total + 255) / 256, 256, 0, stream>>>(x, ei, ea, agg1);
  }
  // conv1 GEMM: 3125*8 tiles, 4 waves/block
  {
    int tiles = (NN / 16) * (HID / 16);
    graphconv_gemm<DIN, true, false><<<tiles / 4, 128, 0, stream>>>(
        agg1, x, Wrel1, Wroot1, brel1, nullptr, h1);
  }
  // conv3 scatter: NE * 32 chunk-threads
  {
    int total = NE * (HID / 4);
    scatter_edges<HID><<<(total + 255) / 256, 256, 0, stream>>>(h1, ei, ea, agg2);
  }
  // conv3 GEMM fused with pool-sum accumulation
  {
    int tiles = (NN / 16) * (HID / 16);
    graphconv_gemm<HID, false, true><<<tiles / 4, 128, 0, stream>>>(
        agg2, h1, Wrel3, Wroot3, brel3, batch, pool);
  }
  count_nodes<<<(NN + 255) / 256, 256, 0, stream>>>(batch, cnt);
  pool_linear<<<NG / 4, 128, 0, stream>>>(pool, cnt, Wlin, blin, out);
  (void)in_sizes; (void)n_in; (void)out_size; (void)ws_size;
}